// GCN_8246337208549
// MI455X (gfx1250) — compile-verified
//
#include <hip/hip_runtime.h>

typedef __attribute__((ext_vector_type(2))) float v2f;
typedef __attribute__((ext_vector_type(8))) float v8f;

#define NNODES_IN 128
#define KDIM 128

// ---------------------------------------------------------------------------
// GEMM: C[M x Ncol] = A[M x 128] * B[128 x Ncol], fp32 via V_WMMA_F32_16X16X4_F32
// Block = 256 threads = 8 waves. Whole B staged in LDS in fragment order.
// JT = number of 16-wide column tiles (8 for Ncol=128, 3 for Ncol=40 padded to 48).
// ---------------------------------------------------------------------------
template <int JT>
__global__ __launch_bounds__(256) void gcn_gemm_wmma(
    const float* __restrict__ A, const float* __restrict__ B,
    float* __restrict__ C, int M, int Ncol) {
  __shared__ float lds[32 * JT * 32 * 2];  // [kb][j][lane][u]

  const int tid = threadIdx.x;
  // Cooperative LDS fill: fragment layout so each wave reads one ds_load_b64 per (kb,j).
  const int total = 32 * JT * 64;
  for (int e = tid; e < total; e += 256) {
    const int u    = e & 1;
    const int lane = (e >> 1) & 31;
    const int t    = e >> 6;
    const int j    = t % JT;
    const int kb   = t / JT;
    const int k    = kb * 4 + ((lane < 16) ? u : (u + 2));
    const int col  = j * 16 + (lane & 15);
    lds[e] = (col < Ncol) ? B[k * Ncol + col] : 0.0f;
  }
  __syncthreads();

  const int wave = tid >> 5;
  const int lane = tid & 31;
  const int tile = blockIdx.x * 8 + wave;           // 16-row tile index
  if (tile * 16 >= M) return;                       // whole-wave exit, EXEC stays full

  const int   row  = tile * 16 + (lane & 15);
  const int   koff = (lane < 16) ? 0 : 2;
  const float* __restrict__ arow = A + (long)row * KDIM;

  v8f acc[JT];
#pragma unroll
  for (int j = 0; j < JT; ++j) acc[j] = (v8f){0.f,0.f,0.f,0.f,0.f,0.f,0.f,0.f};

#pragma unroll 4
  for (int kb = 0; kb < 32; ++kb) {
    const int k = kb * 4 + koff;
    v2f a;
    a.x = arow[k];
    a.y = arow[k + 1];
#pragma unroll
    for (int j = 0; j < JT; ++j) {
      const v2f b = *(const v2f*)&lds[((kb * JT + j) * 32 + lane) * 2];
      acc[j] = __builtin_amdgcn_wmma_f32_16x16x4_f32(
          false, a, false, b, (short)0, acc[j], false, false);
    }
  }

  // C/D layout: VGPR i -> (M = i or i+8, N = lane&15)
  const int baserow = tile * 16 + ((lane < 16) ? 0 : 8);
  const int ccol    = lane & 15;
#pragma unroll
  for (int j = 0; j < JT; ++j) {
    const int col = j * 16 + ccol;
    if (col < Ncol) {
#pragma unroll
      for (int i = 0; i < 8; ++i)
        C[(long)(baserow + i) * Ncol + col] = acc[j][i];
    }
  }
}

// ---------------------------------------------------------------------------
// agg[n][c] = bias[c]
// ---------------------------------------------------------------------------
__global__ void gcn_init_bias(float* __restrict__ agg,
                              const float* __restrict__ bias, int n, int C) {
  const int i = blockIdx.x * blockDim.x + threadIdx.x;
  if (i < n * C) agg[i] = bias[i % C];
}

// ---------------------------------------------------------------------------
// Weighted scatter-add: agg[dst[e]] += T[src[e]] * ew[e]   (float4 chunks)
// ---------------------------------------------------------------------------
__global__ void gcn_scatter_add(const float* __restrict__ T,
                                const int* __restrict__ src,
                                const int* __restrict__ dst,
                                const float* __restrict__ ew,
                                float* __restrict__ agg,
                                int E, int C, int chunks) {
  const int gid = blockIdx.x * blockDim.x + threadIdx.x;
  if (gid >= E * chunks) return;
  const int e = gid / chunks;
  const int q = gid - e * chunks;
  const int s = src[e];
  const int d = dst[e];
  const float w = ew[e];
  const float4 v = ((const float4*)(T + (long)s * C))[q];
  float* out = agg + (long)d * C + q * 4;
  atomicAdd(out + 0, v.x * w);
  atomicAdd(out + 1, v.y * w);
  atomicAdd(out + 2, v.z * w);
  atomicAdd(out + 3, v.w * w);
}

// ---------------------------------------------------------------------------
__global__ void gcn_zero(float* __restrict__ p, int n) {
  const int i = blockIdx.x * blockDim.x + threadIdx.x;
  if (i < n) p[i] = 0.0f;
}

// Per-channel sum / sumsq (C = 128, power of two). Grid stride is a multiple
// of C, so each thread's channel is invariant -> register accumulation.
__global__ __launch_bounds__(256) void gcn_bn_stats(
    const float* __restrict__ H, float* __restrict__ stats, int n, int C) {
  __shared__ float ssum[128];
  __shared__ float ssq[128];
  const int tid = threadIdx.x;
  if (tid < 128) { ssum[tid] = 0.f; ssq[tid] = 0.f; }
  __syncthreads();

  const int c = tid & (C - 1);
  float s = 0.f, s2 = 0.f;
  const long total  = (long)n * C;
  const long stride = (long)gridDim.x * blockDim.x;   // 512*256, multiple of 128
  for (long i = (long)blockIdx.x * blockDim.x + tid; i < total; i += stride) {
    const float v = H[i];
    s += v;
    s2 += v * v;
  }
  atomicAdd(&ssum[c], s);
  atomicAdd(&ssq[c], s2);
  __syncthreads();
  if (tid < C) {
    atomicAdd(&stats[tid],     ssum[tid]);
    atomicAdd(&stats[C + tid], ssq[tid]);
  }
}

// BN (training stats) + ReLU + deterministic hash dropout (p = 0.5, scale 2x)
__global__ void gcn_bn_relu_drop(const float* __restrict__ H,
                                 const float* __restrict__ stats,
                                 const float* __restrict__ gamma,
                                 const float* __restrict__ beta,
                                 float* __restrict__ out,
                                 int n, int C, unsigned seed) {
  const long i = (long)blockIdx.x * blockDim.x + threadIdx.x;
  if (i >= (long)n * C) return;
  const int c = (int)(i & (C - 1));
  const float invN = 1.0f / (float)n;
  const float mean = stats[c] * invN;
  const float var  = stats[C + c] * invN - mean * mean;
  float v = (H[i] - mean) * rsqrtf(var + 1e-5f) * gamma[c] + beta[c];
  v = fmaxf(v, 0.0f);
  unsigned h = (unsigned)i * 2654435761u ^ seed;
  h ^= h >> 16; h *= 2246822519u; h ^= h >> 13; h *= 3266489917u; h ^= h >> 16;
  out[i] = (h & 1u) ? (v * 2.0f) : 0.0f;
}

// ---------------------------------------------------------------------------
// log_softmax over C=40 columns, one wave32 per row.
// ---------------------------------------------------------------------------
__global__ __launch_bounds__(256) void gcn_log_softmax(
    const float* __restrict__ H, float* __restrict__ out, int n, int C) {
  const int row  = blockIdx.x * (blockDim.x >> 5) + (threadIdx.x >> 5);
  const int lane = threadIdx.x & 31;
  if (row >= n) return;
  const float* r = H + (long)row * C;

  float m = -3.0e38f;
  for (int c = lane; c < C; c += 32) m = fmaxf(m, r[c]);
  for (int off = 16; off > 0; off >>= 1) m = fmaxf(m, __shfl_down(m, off));
  m = __shfl(m, 0);

  float s = 0.0f;
  for (int c = lane; c < C; c += 32) s += __expf(r[c] - m);
  for (int off = 16; off > 0; off >>= 1) s += __shfl_down(s, off);
  s = __shfl(s, 0);

  const float lse = m + __logf(s);
  for (int c = lane; c < C; c += 32) out[(long)row * C + c] = r[c] - lse;
}

// ---------------------------------------------------------------------------
extern "C" void kernel_launch(void* const* d_in, const int* in_sizes, int n_in,
                              void* d_out, int out_size, void* d_ws, size_t ws_size,
                              hipStream_t stream) {
  (void)n_in; (void)ws_size;
  const float* x  = (const float*)d_in[0];
  const int*   ei = (const int*)d_in[1];
  const float* ew = (const float*)d_in[2];
  const float* W1 = (const float*)d_in[3];
  const float* b1 = (const float*)d_in[4];
  const float* g1 = (const float*)d_in[5];
  const float* be1= (const float*)d_in[6];
  const float* W2 = (const float*)d_in[7];
  const float* b2 = (const float*)d_in[8];
  const float* g2 = (const float*)d_in[9];
  const float* be2= (const float*)d_in[10];
  const float* W3 = (const float*)d_in[11];
  const float* b3 = (const float*)d_in[12];
  float* out = (float*)d_out;

  const int N = in_sizes[0] / 128;   // 50000
  const int E = in_sizes[2];         // 800000
  const int H = 128;
  const int OUTC = in_sizes[12];     // 40
  const int* src = ei;
  const int* dst = ei + E;

  char* ws   = (char*)d_ws;
  float* t   = (float*)(ws);                         // N*128
  float* ag  = (float*)(ws + (size_t)N * H * 4);     // N*128
  float* hb  = (float*)(ws + (size_t)2 * N * H * 4); // N*128
  float* st  = (float*)(ws + (size_t)3 * N * H * 4); // 256 floats

  const int tilesM     = (N + 15) / 16;              // 3125
  const int gemmBlocks = (tilesM + 7) / 8;           // 391
  const int elemB128   = (N * H + 255) / 256;
  const int scat128    = (E * 32 + 255) / 256;
  const int chunksO    = OUTC / 4;                   // 10
  const int scatO      = (E * chunksO + 255) / 256;
  const int elemBO     = (N * OUTC + 255) / 256;
  const int lsmBlocks  = (N + 7) / 8;

  // ---- Layer 1 ----
  gcn_gemm_wmma<8><<<gemmBlocks, 256, 0, stream>>>(x, W1, t, N, H);
  gcn_init_bias<<<elemB128, 256, 0, stream>>>(ag, b1, N, H);
  gcn_scatter_add<<<scat128, 256, 0, stream>>>(t, src, dst, ew, ag, E, H, 32);
  gcn_zero<<<1, 256, 0, stream>>>(st, 256);
  gcn_bn_stats<<<512, 256, 0, stream>>>(ag, st, N, H);
  gcn_bn_relu_drop<<<elemB128, 256, 0, stream>>>(ag, st, g1, be1, hb, N, H, 0x9E3779B9u);

  // ---- Layer 2 ----
  gcn_gemm_wmma<8><<<gemmBlocks, 256, 0, stream>>>(hb, W2, t, N, H);
  gcn_init_bias<<<elemB128, 256, 0, stream>>>(ag, b2, N, H);
  gcn_scatter_add<<<scat128, 256, 0, stream>>>(t, src, dst, ew, ag, E, H, 32);
  gcn_zero<<<1, 256, 0, stream>>>(st, 256);
  gcn_bn_stats<<<512, 256, 0, stream>>>(ag, st, N, H);
  gcn_bn_relu_drop<<<elemB128, 256, 0, stream>>>(ag, st, g2, be2, hb, N, H, 0x85EBCA6Bu);

  // ---- Layer 3 + log_softmax ----
  gcn_gemm_wmma<3><<<gemmBlocks, 256, 0, stream>>>(hb, W3, t, N, OUTC);
  gcn_init_bias<<<elemBO, 256, 0, stream>>>(ag, b3, N, OUTC);
  gcn_scatter_add<<<scatO, 256, 0, stream>>>(t, src, dst, ew, ag, E, OUTC, chunksO);
  gcn_log_softmax<<<lsmBlocks, 256, 0, stream>>>(ag, out, N, OUTC);
  (void)out_size;
}